// RandSample_1400159339076
// MI455X (gfx1250) — compile-verified
//
#include <hip/hip_runtime.h>
#include <hip/hip_bf16.h>
#include <float.h>

// ---------------- fixed problem dims (from reference) ----------------
#define B_DIM   4
#define N_DIM   16384
#define CIN     64
#define COUT    128
#define M_DIM   4096
#define KNN_K   16
#define EPS_BN  1e-5f

typedef __attribute__((ext_vector_type(2))) float v2f;
typedef __attribute__((ext_vector_type(8))) float v8f;

#if defined(__AMDGCN__)
#  if __has_builtin(__builtin_amdgcn_wmma_f32_16x16x4_f32)
#    define USE_WMMA_F32 1
#  else
#    define USE_WMMA_F32 0
#  endif
#else
#  define USE_WMMA_F32 0
#endif

// =====================================================================
// Kernel 1: z = relu(bn(x @ W^T))  -> mlp  [(B*N) x COUT], fp32 WMMA
//   grid.x = (B*N)/16 row tiles, block = 256 threads = 8 waves,
//   wave w owns column tile w (16 channels).
//   A lane layout (16x4 f32): row = lane&15, K-pair = {0,1} or {2,3} by lane>=16
//   B lane layout (4x16 f32): col = lane&15, same K-pair split
//   C/D: vgpr v holds M = v + 8*(lane>=16), N = lane&15
// =====================================================================
__global__ __launch_bounds__(256)
void gemm_bn_relu_kernel(const float* __restrict__ x,   // (B*N, CIN)
                         const float* __restrict__ W,   // (COUT, CIN)
                         const float* __restrict__ gamma,
                         const float* __restrict__ beta,
                         const float* __restrict__ mean,
                         const float* __restrict__ var,
                         float* __restrict__ mlp)       // (B*N, COUT)
{
    const int lane    = threadIdx.x & 31;
    const int wave    = threadIdx.x >> 5;      // 0..7 -> column tile
    const int rowTile = blockIdx.x;            // 0..(B*N/16 - 1)
    const int r       = lane & 15;
    const int hi      = lane >> 4;             // 0 or 1 (K-pair select)

    const int rowA = rowTile * 16 + r;         // A row this lane feeds
    const int c    = wave * 16 + (lane & 15);  // output channel this lane owns

    const float* __restrict__ xrow = x + (size_t)rowA * CIN + 2 * hi;
    const float* __restrict__ wrow = W + (size_t)c    * CIN + 2 * hi;

    v8f acc = {};

#if USE_WMMA_F32
    #pragma unroll
    for (int kk = 0; kk < CIN / 4; ++kk) {
        v2f a, bb;
        a.x  = xrow[kk * 4 + 0];
        a.y  = xrow[kk * 4 + 1];
        bb.x = wrow[kk * 4 + 0];
        bb.y = wrow[kk * 4 + 1];
        acc = __builtin_amdgcn_wmma_f32_16x16x4_f32(
            false, a, false, bb, (short)0, acc, false, false);
    }
#else
    // Scalar fallback producing the identical accumulator layout:
    // acc[v] = C[M][N] with M = v + 8*hi, N = c (this lane's channel).
    #pragma unroll
    for (int v = 0; v < 8; ++v) {
        const int mrow = rowTile * 16 + v + 8 * hi;
        const float* __restrict__ xr = x + (size_t)mrow * CIN;
        const float* __restrict__ wr = W + (size_t)c    * CIN;
        float s = 0.f;
        for (int k = 0; k < CIN; ++k) s = fmaf(xr[k], wr[k], s);
        acc[v] = s;
    }
#endif

    // BN + ReLU epilogue: one channel per lane.
    const float g   = gamma[c];
    const float bt  = beta[c];
    const float mn  = mean[c];
    const float inv = rsqrtf(var[c] + EPS_BN);

    float* __restrict__ outb = mlp + (size_t)(rowTile * 16) * COUT + c;
    #pragma unroll
    for (int v = 0; v < 8; ++v) {
        const int mrow = v + 8 * hi;
        float z = fmaf(g * (acc[v] - mn), inv, bt);
        z = fmaxf(z, 0.f);
        outb[(size_t)mrow * COUT] = z;
    }
}

// =====================================================================
// Kernel 2: brute-force KNN (top-16 smallest d2) + p_out gather.
//   256 threads/block = 32 queries x 8 splits. Split s handles points
//   with index % 8 == s. Point tile staged through LDS. Per-thread
//   register top-16 (replace-worst), then LDS merge per query.
// =====================================================================
#define KNN_TILE 1024
#define KNN_QPB  32

__global__ __launch_bounds__(256)
void knn_kernel(const float* __restrict__ p,          // (B, N, 3)
                const int*   __restrict__ sample_idx, // (M)
                int*         __restrict__ nbrs,       // (B*M, 16)
                float*       __restrict__ p_out)      // (B*M, 3)
{
    __shared__ float spx[KNN_TILE];
    __shared__ float spy[KNN_TILE];
    __shared__ float spz[KNN_TILE];
    __shared__ float s_md[256 * KNN_K];
    __shared__ int   s_mi[256 * KNN_K];

    const int t  = threadIdx.x;
    const int ql = t >> 3;               // query within block 0..31
    const int s  = t & 7;                // split 0..7
    const int q  = blockIdx.x * KNN_QPB + ql;   // global query 0..B*M-1
    const int b  = q / M_DIM;
    const int m  = q - b * M_DIM;

    const int   sidx  = sample_idx[m];
    const float* qptr = p + ((size_t)b * N_DIM + sidx) * 3;
    const float qx = qptr[0], qy = qptr[1], qz = qptr[2];

    if (s == 0) {
        p_out[(size_t)q * 3 + 0] = qx;
        p_out[(size_t)q * 3 + 1] = qy;
        p_out[(size_t)q * 3 + 2] = qz;
    }

    float best_d[KNN_K];
    int   best_i[KNN_K];
    #pragma unroll
    for (int j = 0; j < KNN_K; ++j) { best_d[j] = FLT_MAX; best_i[j] = 0; }
    float worst = FLT_MAX;
    int   wslot = 0;

    const float* __restrict__ pb = p + (size_t)b * N_DIM * 3;

    for (int base = 0; base < N_DIM; base += KNN_TILE) {
        // cooperative load of KNN_TILE points into LDS
        for (int i = t; i < KNN_TILE; i += 256) {
            const float* pp = pb + (size_t)(base + i) * 3;
            spx[i] = pp[0]; spy[i] = pp[1]; spz[i] = pp[2];
        }
        __syncthreads();

        for (int i = s; i < KNN_TILE; i += 8) {
            const float dx = qx - spx[i];
            const float dy = qy - spy[i];
            const float dz = qz - spz[i];
            const float d  = fmaf(dx, dx, fmaf(dy, dy, dz * dz));
            if (d < worst) {
                const int gi = base + i;
                #pragma unroll
                for (int j = 0; j < KNN_K; ++j)
                    if (j == wslot) { best_d[j] = d; best_i[j] = gi; }
                worst = best_d[0]; wslot = 0;
                #pragma unroll
                for (int j = 1; j < KNN_K; ++j)
                    if (best_d[j] > worst) { worst = best_d[j]; wslot = j; }
            }
        }
        __syncthreads();
    }

    // dump per-thread lists to LDS
    #pragma unroll
    for (int j = 0; j < KNN_K; ++j) {
        s_md[t * KNN_K + j] = best_d[j];
        s_mi[t * KNN_K + j] = best_i[j];
    }
    __syncthreads();

    // one thread per query merges its 8 lists (128 entries -> 16 minima)
    if (s == 0) {
        const int base = (ql * 8) * KNN_K;       // 128 entries
        int* __restrict__ dst = nbrs + (size_t)q * KNN_K;
        for (int r = 0; r < KNN_K; ++r) {
            float md = FLT_MAX;
            int   mi = 0, mp = base;
            for (int e = 0; e < 8 * KNN_K; ++e) {
                const float d = s_md[base + e];
                if (d < md) { md = d; mi = s_mi[base + e]; mp = base + e; }
            }
            s_md[mp] = FLT_MAX;
            dst[r] = mi;
        }
    }
}

// =====================================================================
// Kernel 3: y[q][c] = max_k mlp[b*N + nbrs[q][k]][c]
//   32 threads/query, one float4 channel slice each.
// =====================================================================
__global__ __launch_bounds__(256)
void gather_max_kernel(const float* __restrict__ mlp,  // (B*N, COUT)
                       const int*   __restrict__ nbrs, // (B*M, 16)
                       float*       __restrict__ y)    // (B*M, COUT)
{
    const int t  = threadIdx.x;
    const int q  = blockIdx.x * 8 + (t >> 5);   // query
    const int c4 = t & 31;                      // float4 slice 0..31
    const int b  = q / M_DIM;

    const int* __restrict__ nb = nbrs + (size_t)q * KNN_K;
    float4 mx = make_float4(-FLT_MAX, -FLT_MAX, -FLT_MAX, -FLT_MAX);
    #pragma unroll
    for (int k = 0; k < KNN_K; ++k) {
        const int n = nb[k];
        const float4 v = *reinterpret_cast<const float4*>(
            mlp + ((size_t)b * N_DIM + n) * COUT + c4 * 4);
        mx.x = fmaxf(mx.x, v.x);
        mx.y = fmaxf(mx.y, v.y);
        mx.z = fmaxf(mx.z, v.z);
        mx.w = fmaxf(mx.w, v.w);
    }
    *reinterpret_cast<float4*>(y + (size_t)q * COUT + c4 * 4) = mx;
}

// =====================================================================
extern "C" void kernel_launch(void* const* d_in, const int* in_sizes, int n_in,
                              void* d_out, int out_size, void* d_ws, size_t ws_size,
                              hipStream_t stream) {
    const float* x          = (const float*)d_in[0];  // (B,N,CIN)
    const float* p          = (const float*)d_in[1];  // (B,N,3)
    const int*   sample_idx = (const int*)  d_in[2];  // (M)
    const float* W          = (const float*)d_in[3];  // (COUT,CIN)
    const float* gamma      = (const float*)d_in[4];
    const float* beta       = (const float*)d_in[5];
    const float* mean       = (const float*)d_in[6];
    const float* var        = (const float*)d_in[7];

    float* y     = (float*)d_out;                                   // (B*M, COUT)
    float* p_out = (float*)d_out + (size_t)B_DIM * M_DIM * COUT;    // (B*M, 3)

    float* mlp  = (float*)d_ws;                                     // (B*N, COUT)
    int*   nbrs = (int*)((char*)d_ws +
                         (size_t)B_DIM * N_DIM * COUT * sizeof(float)); // (B*M, 16)

    // 1) GEMM + BN + ReLU  (WMMA f32 16x16x4)
    {
        const int rowTiles = (B_DIM * N_DIM) / 16;   // 4096
        gemm_bn_relu_kernel<<<rowTiles, 256, 0, stream>>>(
            x, W, gamma, beta, mean, var, mlp);
    }
    // 2) KNN + p_out gather (independent of 1, stream-ordered anyway)
    {
        const int blocks = (B_DIM * M_DIM) / KNN_QPB; // 512
        knn_kernel<<<blocks, 256, 0, stream>>>(p, sample_idx, nbrs, p_out);
    }
    // 3) gather + max over K
    {
        const int blocks = (B_DIM * M_DIM) / 8;       // 2048
        gather_max_kernel<<<blocks, 256, 0, stream>>>(mlp, nbrs, y);
    }
}